// SCELoss_13056700580270
// MI455X (gfx1250) — compile-verified
//
#include <hip/hip_runtime.h>
#include <hip/hip_bf16.h>
#include <math.h>
#include <stdint.h>

typedef float v2f __attribute__((ext_vector_type(2)));
typedef float v8f __attribute__((ext_vector_type(8)));

#define WAVE 32
#define BLOCK 256

// Full-wave sum of (a0 + a1) over all 32 lanes using V_WMMA_F32_16X16X4_F32
// with B = ones: D[m][n] = sum_k A[m][k], i.e. row sums of the 16x4 A tile.
// A layout (32-bit A, 16x4): lanes 0-15 hold (M=lane, K=0/1) in the two A
// VGPRs, lanes 16-31 hold (M=lane-16, K=2/3).  So row sums already combine
// lane L and lane L+16.  D layout: lane n, vgpr j -> D[M = j (+8 for hi
// lanes)][N = n]; summing the 8 D vgprs per lane and adding lanes n / n+16
// yields the total of all 64 A elements in every lane.
__device__ __forceinline__ float wave_sum64_wmma(float a0, float a1) {
  v2f A; A[0] = a0;   A[1] = a1;
  v2f B; B[0] = 1.0f; B[1] = 1.0f;
  v8f C = {0.f, 0.f, 0.f, 0.f, 0.f, 0.f, 0.f, 0.f};
  v8f D = __builtin_amdgcn_wmma_f32_16x16x4_f32(
      /*neg_a=*/false, A, /*neg_b=*/false, B,
      /*c_mod=*/(short)0, C, /*reuse_a=*/false, /*reuse_b=*/false);
  float p = D[0] + D[1] + D[2] + D[3] + D[4] + D[5] + D[6] + D[7];
  p += __shfl_xor(p, 16, WAVE);
  return p;
}

__device__ __forceinline__ float wave_max(float v) {
#pragma unroll
  for (int m = WAVE / 2; m >= 1; m >>= 1)
    v = fmaxf(v, __shfl_xor(v, m, WAVE));
  return v;
}

__global__ void __launch_bounds__(BLOCK) sce_zero(float* out) { out[0] = 0.0f; }

// Pass 1: online softmax (max + sumexp in ONE HBM read), writes M,Z per row
// and atomically accumulates the CE term:  ce = log Z + M - x[label].
__global__ void __launch_bounds__(BLOCK) sce_pass1(
    const float* __restrict__ pred, const int* __restrict__ labels,
    int V, float inv_n,
    float* __restrict__ rowM, float* __restrict__ rowZ, float* __restrict__ out)
{
  const int row  = blockIdx.x;
  const float* x = pred + (size_t)row * (size_t)V;
  const int tid  = threadIdx.x;
  const int nthr = blockDim.x;

  float m = -INFINITY, s0 = 0.f, s1 = 0.f;

  // Row base may be misaligned (V odd): peel to 16B, float4 body, scalar tail.
  int pad = (int)((4u - (((uintptr_t)x >> 2) & 3u)) & 3u);
  if (pad > V) pad = V;
  for (int i = tid; i < pad; i += nthr) {
    float v = x[i];
    if (v > m) { float sc = __expf(m - v); s0 *= sc; s1 *= sc; m = v; }
    s0 += __expf(v - m);
  }
  const float4* xv = (const float4*)(x + pad);
  const int nvec = (V - pad) >> 2;
  for (int i = tid; i < nvec; i += nthr) {
    float4 v = xv[i];
    __builtin_prefetch(xv + i + nthr, 0, 1);     // global_prefetch_b8
    float vm = fmaxf(fmaxf(v.x, v.y), fmaxf(v.z, v.w));
    if (vm > m) { float sc = __expf(m - vm); s0 *= sc; s1 *= sc; m = vm; }
    s0 += __expf(v.x - m) + __expf(v.y - m);
    s1 += __expf(v.z - m) + __expf(v.w - m);
  }
  for (int i = pad + (nvec << 2) + tid; i < V; i += nthr) {
    float v = x[i];
    if (v > m) { float sc = __expf(m - v); s0 *= sc; s1 *= sc; m = v; }
    s0 += __expf(v - m);
  }

  // Wave: max, rescale to wave max, WMMA sum (EXEC all ones here).
  float wm   = wave_max(m);
  float sc   = __expf(m - wm);
  float wsum = wave_sum64_wmma(s0 * sc, s1 * sc);

  __shared__ float smM[BLOCK / WAVE], smS[BLOCK / WAVE];
  const int wid = tid >> 5, lane = tid & 31;
  if (lane == 0) { smM[wid] = wm; smS[wid] = wsum; }
  __syncthreads();
  if (tid == 0) {
    const int nw = nthr >> 5;
    float M = -INFINITY;
    for (int w = 0; w < nw; ++w) M = fmaxf(M, smM[w]);
    float Z = 0.f;
    for (int w = 0; w < nw; ++w) Z += smS[w] * __expf(smM[w] - M);
    rowM[row] = M;
    rowZ[row] = Z;
    float xlab = x[labels[row]];
    float ce = __logf(Z) + M - xlab;            // -log_softmax[label]
    atomicAdd(out, ce * inv_n);                 // ALPHA = 1
  }
}

// Pass 2: second HBM read computes sum of clamp(p, 1e-7, 1) in exp-domain
// (threshold 1e-7*Z, cap Z -> one divide by Z per row), accumulates RCE.
__global__ void __launch_bounds__(BLOCK) sce_pass2(
    const float* __restrict__ pred, const int* __restrict__ labels,
    int V, float inv_n,
    const float* __restrict__ rowM, const float* __restrict__ rowZ,
    float* __restrict__ out)
{
  const int row  = blockIdx.x;
  const float* x = pred + (size_t)row * (size_t)V;
  const int tid  = threadIdx.x;
  const int nthr = blockDim.x;
  const float M  = rowM[row];
  const float Z  = rowZ[row];
  const float lo = 1e-7f * Z;
  const float hi = Z;

  float s0 = 0.f, s1 = 0.f;
  int pad = (int)((4u - (((uintptr_t)x >> 2) & 3u)) & 3u);
  if (pad > V) pad = V;
  for (int i = tid; i < pad; i += nthr)
    s0 += fminf(fmaxf(__expf(x[i] - M), lo), hi);
  const float4* xv = (const float4*)(x + pad);
  const int nvec = (V - pad) >> 2;
  for (int i = tid; i < nvec; i += nthr) {
    float4 v = xv[i];
    __builtin_prefetch(xv + i + nthr, 0, 1);
    s0 += fminf(fmaxf(__expf(v.x - M), lo), hi)
        + fminf(fmaxf(__expf(v.y - M), lo), hi);
    s1 += fminf(fmaxf(__expf(v.z - M), lo), hi)
        + fminf(fmaxf(__expf(v.w - M), lo), hi);
  }
  for (int i = pad + (nvec << 2) + tid; i < V; i += nthr)
    s0 += fminf(fmaxf(__expf(x[i] - M), lo), hi);

  float wsum = wave_sum64_wmma(s0, s1);

  __shared__ float smS[BLOCK / WAVE];
  const int wid = tid >> 5, lane = tid & 31;
  if (lane == 0) smS[wid] = wsum;
  __syncthreads();
  if (tid == 0) {
    const int nw = nthr >> 5;
    float S = 0.f;
    for (int w = 0; w < nw; ++w) S += smS[w];
    float psum = S / Z;                                        // sum clamp(p)
    float plab = fminf(fmaxf(__expf(x[labels[row]] - M) / Z, 1e-7f), 1.0f);
    const float neg_log_min = 9.210340371976184f;              // -log(1e-4)
    float rce = neg_log_min * (psum - plab);
    atomicAdd(out, rce * inv_n);                               // BETA = 1
  }
}

extern "C" void kernel_launch(void* const* d_in, const int* in_sizes, int n_in,
                              void* d_out, int out_size, void* d_ws, size_t ws_size,
                              hipStream_t stream) {
  const float* pred  = (const float*)d_in[0];
  const int* labels  = (const int*)d_in[1];
  const int nrows    = in_sizes[1];
  const int V        = in_sizes[0] / nrows;
  float* out         = (float*)d_out;
  float* rowM        = (float*)d_ws;
  float* rowZ        = rowM + nrows;
  const float inv_n  = 1.0f / (float)nrows;

  sce_zero <<<1, BLOCK, 0, stream>>>(out);
  sce_pass1<<<nrows, BLOCK, 0, stream>>>(pred, labels, V, inv_n, rowM, rowZ, out);
  sce_pass2<<<nrows, BLOCK, 0, stream>>>(pred, labels, V, inv_n, rowM, rowZ, out);
}